// Yolov8_82557861363908
// MI455X (gfx1250) — compile-verified
//
#include <hip/hip_runtime.h>
#include <math.h>
#include <float.h>

typedef float v2f __attribute__((ext_vector_type(2)));
typedef float v8f __attribute__((ext_vector_type(8)));

#define NB 32
#define NA 8400
#define NCH 144
#define NCLS 80
#define TOPK 1000
#define MAXDET 300
#define SORTN 16384
#define CONF_T 0.25f
#define IOU_T 0.45f
#define MAXWH 7680.0f

// ---------- helpers ----------

__device__ __forceinline__ unsigned int ford(float f) {
    unsigned int u = __float_as_uint(f);
    return (u & 0x80000000u) ? ~u : (u | 0x80000000u);
}
// key sorts ASCENDING == score DESCENDING, ties broken by ascending index (stable)
__device__ __forceinline__ unsigned long long dkey(float s, unsigned int idx) {
    unsigned int o = ~ford(s);
    return (((unsigned long long)o) << 32) | (unsigned long long)idx;
}

__device__ __forceinline__ void anchor_of(int a, float& ax, float& ay, float& s) {
    int idx, w;
    if (a < 6400)      { idx = a;        w = 80; s = 8.0f;  }
    else if (a < 8000) { idx = a - 6400; w = 40; s = 16.0f; }
    else               { idx = a - 8000; w = 20; s = 32.0f; }
    ax = (float)(idx % w) + 0.5f;
    ay = (float)(idx / w) + 0.5f;
}

// single-block bitonic sort, ascending by 64-bit key; works on LDS or global.
__device__ __forceinline__ void bitonic_sort(unsigned long long* k, int n, int tid, int nth) {
    for (int sz = 2; sz <= n; sz <<= 1) {
        for (int st = sz >> 1; st > 0; st >>= 1) {
            __syncthreads();
            for (int i = tid; i < n; i += nth) {
                int j = i ^ st;
                if (j > i) {
                    bool up = ((i & sz) == 0);
                    unsigned long long a = k[i], b = k[j];
                    if ((a > b) == up) { k[i] = b; k[j] = a; }
                }
            }
        }
    }
    __syncthreads();
}

// ---------- kernel 1: DFL decode via fp32 WMMA ----------
// Each wave handles 4 anchors (16 (q,anchor) items). Lanes L and L+16 both own
// item L; lane-lo supplies K%4 in {0,1}, lane-hi supplies K%4 in {2,3}, per the
// 32-bit A/B 16x4 WMMA layouts. A = broadcast dfl weights (all rows equal), so
// D(m, n) = sum_k w_k * P(k, n) for every m -> acc[0] holds dist(item = lane&15).
__global__ void __launch_bounds__(256) yolo_dfl_wmma(const float* __restrict__ preds,
                                                     const float* __restrict__ dflw,
                                                     float* __restrict__ boxes) {
    int wave = blockIdx.x * 8 + (threadIdx.x >> 5);
    int lane = threadIdx.x & 31;
    int b   = wave / (NA / 4);
    int grp = wave - b * (NA / 4);
    int a0  = grp * 4;

    int lo = lane & 15;
    int hi = lane >> 4;          // 0 or 1
    int q  = lo & 3;             // box side 0..3
    int al = lo >> 2;            // anchor within group
    int a  = a0 + al;

    // 16 bin logits for this (b, q, a)
    size_t base = ((size_t)b * NCH + (size_t)q * 16) * NA + (size_t)a;
    float x[16];
#pragma unroll
    for (int r = 0; r < 16; ++r) x[r] = preds[base + (size_t)r * NA];

    float mx = x[0];
#pragma unroll
    for (int r = 1; r < 16; ++r) mx = fmaxf(mx, x[r]);
    float p[16];
    float sum = 0.0f;
#pragma unroll
    for (int r = 0; r < 16; ++r) { p[r] = expf(x[r] - mx); sum += p[r]; }
    float inv = 1.0f / sum;
#pragma unroll
    for (int r = 0; r < 16; ++r) p[r] *= inv;

    float wv[16];
#pragma unroll
    for (int r = 0; r < 16; ++r) wv[r] = dflw[r];

    v8f acc = {0.0f, 0.0f, 0.0f, 0.0f, 0.0f, 0.0f, 0.0f, 0.0f};
#pragma unroll
    for (int g = 0; g < 4; ++g) {
        float a0v = hi ? wv[4 * g + 2] : wv[4 * g + 0];
        float a1v = hi ? wv[4 * g + 3] : wv[4 * g + 1];
        float b0v = hi ? p[4 * g + 2]  : p[4 * g + 0];
        float b1v = hi ? p[4 * g + 3]  : p[4 * g + 1];
        v2f av; av[0] = a0v; av[1] = a1v;
        v2f bv; bv[0] = b0v; bv[1] = b1v;
        acc = __builtin_amdgcn_wmma_f32_16x16x4_f32(false, av, false, bv,
                                                    (short)0, acc, false, false);
    }
    float dist = acc[0];   // dist for item (lane & 15)

    // gather the 4 sides of anchor (lane&3) into one lane
    int an = lane & 3;
    float d0 = __shfl(dist, an * 4 + 0, 32);
    float d1 = __shfl(dist, an * 4 + 1, 32);
    float d2 = __shfl(dist, an * 4 + 2, 32);
    float d3 = __shfl(dist, an * 4 + 3, 32);

    if (lane < 4) {
        int aa = a0 + lane;
        float ax, ay, s;
        anchor_of(aa, ax, ay, s);
        float x1 = ax - d0, y1 = ay - d1;
        float x2 = ax + d2, y2 = ay + d3;
        float cx = (x1 + x2) * 0.5f * s;
        float cy = (y1 + y2) * 0.5f * s;
        float w  = (x2 - x1) * s;
        float h  = (y2 - y1) * s;
        float4 bb;
        bb.x = cx - w * 0.5f;
        bb.y = cy - h * 0.5f;
        bb.z = cx + w * 0.5f;
        bb.w = cy + h * 0.5f;
        *(float4*)(boxes + ((size_t)b * NA + (size_t)aa) * 4) = bb;
    }
}

// ---------- kernel 2: class max/argmax + sigmoid + threshold ----------
__global__ void __launch_bounds__(256) yolo_cls(const float* __restrict__ preds,
                                                float* __restrict__ score,
                                                int* __restrict__ clsix) {
    int tid = blockIdx.x * 256 + threadIdx.x;
    int b = tid / NA;
    int a = tid - b * NA;
    size_t base = ((size_t)b * NCH + 64) * NA + (size_t)a;
    float m = -FLT_MAX;
    int arg = 0;
#pragma unroll 4
    for (int c = 0; c < NCLS; ++c) {
        float v = preds[base + (size_t)c * NA];
        if (v > m) { m = v; arg = c; }   // strict > == first-max (jnp.argmax)
    }
    float conf = 1.0f / (1.0f + expf(-m));
    score[tid] = (conf > CONF_T) ? conf : -1.0f;
    clsix[tid] = arg;
}

// ---------- kernel 3: per-batch top-1000 (full bitonic sort, stable) ----------
__global__ void __launch_bounds__(1024) yolo_topk(const float* __restrict__ score,
                                                  const float* __restrict__ boxes,
                                                  const int* __restrict__ clsix,
                                                  unsigned long long* __restrict__ keys,
                                                  float* __restrict__ cbox,
                                                  float* __restrict__ cobox,
                                                  float* __restrict__ cscore,
                                                  float* __restrict__ ccls) {
    int b = blockIdx.x;
    int tid = threadIdx.x;
    const float* sb = score + (size_t)b * NA;
    unsigned long long* kb = keys + (size_t)b * SORTN;

    for (int i = tid; i < SORTN; i += 1024) {
        kb[i] = (i < NA) ? dkey(sb[i], (unsigned)i) : ~0ull;
    }
    bitonic_sort(kb, SORTN, tid, 1024);

    for (int k = tid; k < TOPK; k += 1024) {
        unsigned long long key = kb[k];
        int idx = (int)(key & 0xffffffffull);
        size_t src = (size_t)b * NA + (size_t)idx;
        float s = sb[idx];
        const float* bx = boxes + src * 4;
        int c = clsix[src];
        float off = (float)c * MAXWH;
        size_t dst = (size_t)b * TOPK + (size_t)k;
        cbox[dst * 4 + 0] = bx[0];
        cbox[dst * 4 + 1] = bx[1];
        cbox[dst * 4 + 2] = bx[2];
        cbox[dst * 4 + 3] = bx[3];
        cobox[dst * 4 + 0] = bx[0] + off;
        cobox[dst * 4 + 1] = bx[1] + off;
        cobox[dst * 4 + 2] = bx[2] + off;
        cobox[dst * 4 + 3] = bx[3] + off;
        cscore[dst] = s;
        ccls[dst]   = (float)c;
    }
}

// ---------- kernel 4: greedy NMS (one block per batch) ----------
__global__ void __launch_bounds__(1024) yolo_nms(const float* __restrict__ cobox,
                                                 const float* __restrict__ cscore,
                                                 int* __restrict__ keep) {
    __shared__ float sx1[TOPK], sy1[TOPK], sx2[TOPK], sy2[TOPK], sar[TOPK], ssc[TOPK];
    __shared__ int skp[TOPK];
    int b = blockIdx.x;
    int tid = threadIdx.x;

    for (int k = tid; k < TOPK; k += 1024) {
        size_t s = (size_t)b * TOPK + (size_t)k;
        float x1 = cobox[s * 4 + 0], y1 = cobox[s * 4 + 1];
        float x2 = cobox[s * 4 + 2], y2 = cobox[s * 4 + 3];
        sx1[k] = x1; sy1[k] = y1; sx2[k] = x2; sy2[k] = y2;
        sar[k] = (x2 - x1) * (y2 - y1);
        ssc[k] = cscore[s];
        skp[k] = 0;
    }
    __syncthreads();

    for (int i = 0; i < TOPK; ++i) {
        float ix1 = sx1[i], iy1 = sy1[i], ix2 = sx2[i], iy2 = sy2[i], iar = sar[i];
        int sup = 0;
        int j = tid;
        if (j < i && skp[j]) {
            float lx = fmaxf(ix1, sx1[j]);
            float ly = fmaxf(iy1, sy1[j]);
            float rx = fminf(ix2, sx2[j]);
            float ry = fminf(iy2, sy2[j]);
            float w = fmaxf(rx - lx, 0.0f);
            float h = fmaxf(ry - ly, 0.0f);
            float inter = w * h;
            float iou = inter / (iar + sar[j] - inter + 1e-7f);
            if (iou > IOU_T) sup = 1;
        }
        int any = __syncthreads_or(sup);
        if (tid == 0) skp[i] = (ssc[i] > CONF_T) && !any;
        __syncthreads();
    }

    for (int k = tid; k < TOPK; k += 1024) keep[(size_t)b * TOPK + k] = skp[k];
}

// ---------- kernel 5: final stable sort + emit 300 x 6 ----------
__global__ void __launch_bounds__(1024) yolo_final(const float* __restrict__ cbox,
                                                   const float* __restrict__ cscore,
                                                   const float* __restrict__ ccls,
                                                   const int* __restrict__ keep,
                                                   float* __restrict__ out) {
    __shared__ unsigned long long fk[1024];
    int b = blockIdx.x;
    int k = threadIdx.x;

    if (k < TOPK) {
        size_t s = (size_t)b * TOPK + (size_t)k;
        float sel = keep[s] ? cscore[s] : -1e9f;
        fk[k] = dkey(sel, (unsigned)k);
    } else {
        fk[k] = ~0ull;
    }
    bitonic_sort(fk, 1024, k, 1024);

    if (k < MAXDET) {
        int idx = (int)(fk[k] & 0xffffffffull);
        size_t s = (size_t)b * TOPK + (size_t)idx;
        float* o = out + ((size_t)b * MAXDET + (size_t)k) * 6;
        if (keep[s]) {
            o[0] = cbox[s * 4 + 0];
            o[1] = cbox[s * 4 + 1];
            o[2] = cbox[s * 4 + 2];
            o[3] = cbox[s * 4 + 3];
            o[4] = cscore[s];
            o[5] = ccls[s];
        } else {
            o[0] = 0.0f; o[1] = 0.0f; o[2] = 0.0f;
            o[3] = 0.0f; o[4] = 0.0f; o[5] = 0.0f;
        }
    }
}

// ---------- host launch ----------
extern "C" void kernel_launch(void* const* d_in, const int* in_sizes, int n_in,
                              void* d_out, int out_size, void* d_ws, size_t ws_size,
                              hipStream_t stream) {
    (void)in_sizes; (void)n_in; (void)out_size; (void)ws_size;
    const float* preds = (const float*)d_in[0];
    const float* dflw  = (const float*)d_in[1];
    float* out = (float*)d_out;

    char* ws = (char*)d_ws;
    size_t off = 0;
    float* boxes = (float*)(ws + off);                 off += (size_t)NB * NA * 4 * sizeof(float);
    float* score = (float*)(ws + off);                 off += (size_t)NB * NA * sizeof(float);
    int*   clsix = (int*)(ws + off);                   off += (size_t)NB * NA * sizeof(int);
    unsigned long long* keys = (unsigned long long*)(ws + off); off += (size_t)NB * SORTN * sizeof(unsigned long long);
    float* cbox   = (float*)(ws + off);                off += (size_t)NB * TOPK * 4 * sizeof(float);
    float* cobox  = (float*)(ws + off);                off += (size_t)NB * TOPK * 4 * sizeof(float);
    float* cscore = (float*)(ws + off);                off += (size_t)NB * TOPK * sizeof(float);
    float* ccls   = (float*)(ws + off);                off += (size_t)NB * TOPK * sizeof(float);
    int*   keep   = (int*)(ws + off);                  off += (size_t)NB * TOPK * sizeof(int);

    // 1) DFL decode (WMMA): 32 batches * 2100 anchor-groups = 67200 waves = 8400 blocks
    yolo_dfl_wmma<<<dim3(NB * (NA / 4) / 8), dim3(256), 0, stream>>>(preds, dflw, boxes);
    // 2) class score: 268800 threads
    yolo_cls<<<dim3(NB * NA / 256), dim3(256), 0, stream>>>(preds, score, clsix);
    // 3) top-1000 per batch
    yolo_topk<<<dim3(NB), dim3(1024), 0, stream>>>(score, boxes, clsix, keys,
                                                   cbox, cobox, cscore, ccls);
    // 4) greedy NMS per batch
    yolo_nms<<<dim3(NB), dim3(1024), 0, stream>>>(cobox, cscore, keep);
    // 5) final ordering + output
    yolo_final<<<dim3(NB), dim3(1024), 0, stream>>>(cbox, cscore, ccls, keep, out);
}